// FasterRCNN_4544075399246
// MI455X (gfx1250) — compile-verified
//
#include <hip/hip_runtime.h>
#include <math.h>

// ---------------------------------------------------------------------------
// Faster-RCNN RPN head + proposal generation for MI455X (gfx1250).
// Conv3x3 (256->256) done as implicit GEMM on V_WMMA_F32_16X16X4_F32 (full
// fp32 matrix path -> ordering-faithful scores), heads fused in-block.
// 32-pixel N-tiles: A (weight) fragments amortized over two B fragments.
// ---------------------------------------------------------------------------

typedef float v2f __attribute__((ext_vector_type(2)));
typedef float v8f __attribute__((ext_vector_type(8)));

#define IMG_W_F   1216.0f
#define IMG_H_F   800.0f
#define BBOX_CLIP 4.135166556742356f   // log(1000/16)
#define LVTOP     1000
#define CAND_CAP  6144

#define TPB  32                         // pixels per block tile
#define XCOL (TPB + 2)                  // 34 staged columns (x0-1 .. x0+32)
#define XROW (3 * XCOL)                 // 102 floats per channel
#define HSTR (TPB + 1)                  // h-tile stride (bank spread)

// ----------------------------- init fills ----------------------------------
__global__ void fill_f32(float* p, int n, float v) {
    int i = blockIdx.x * blockDim.x + threadIdx.x;
    if (i < n) p[i] = v;
}
__global__ void fill_u32(unsigned* p, int n) {
    int i = blockIdx.x * blockDim.x + threadIdx.x;
    if (i < n) p[i] = 0u;
}

// ------------------- fused conv3x3 + relu + 1x1 heads -----------------------
// Block: 256 thr (8 waves). Grid: (ceil(W/32), H, B).
// Each block computes h = relu(conv3x3(x)+b) for a 1x32 pixel strip, all 256
// channels. Each wave owns 32 out-ch x 32 px = 2 M-tiles x 2 N-tiles of
// v_wmma_f32_16x16x4_f32; A fragments are shared across both N-tiles.
// Then the 12-ch reg and 3-ch cls(1x1)+sigmoid heads run off the LDS h tile.
__global__ __launch_bounds__(256)
void conv_head_fused(const float* __restrict__ X,  const float* __restrict__ Wc,
                     const float* __restrict__ Bc, const float* __restrict__ Wcls,
                     const float* __restrict__ Bcls, const float* __restrict__ Wreg,
                     const float* __restrict__ Breg,
                     float* __restrict__ regOut, float* __restrict__ scrOut,
                     int H, int Wd, int N)
{
    __shared__ float Xs[256 * XROW];        // 256 ci x 3 rows x 34 cols (~102 KB)
    const int tid = threadIdx.x;
    const int b  = blockIdx.z;
    const int y  = blockIdx.y;
    const int x0 = blockIdx.x * TPB;

    // stage input patch (zero-padded borders)
    for (int t = tid; t < 256 * XROW; t += 256) {
        int ci = t / XROW, r = t % XROW, ry = r / XCOL, cx = r % XCOL;
        int gy = y + ry - 1, gx = x0 + cx - 1;
        float v = 0.0f;
        if (gy >= 0 && gy < H && gx >= 0 && gx < Wd)
            v = X[(((size_t)b * 256 + ci) * H + gy) * Wd + gx];
        Xs[t] = v;
    }
    __syncthreads();

    const int wave = tid >> 5, lane = tid & 31;
    const int half = lane >> 4, mn = lane & 15;   // mn: M row for A, N col for B
    const int co0 = wave * 32, co1 = co0 + 16;

    v8f acc00 = {0.f,0.f,0.f,0.f,0.f,0.f,0.f,0.f};  // (co0 tile, px 0-15)
    v8f acc01 = acc00;                               // (co0 tile, px 16-31)
    v8f acc10 = acc00;                               // (co1 tile, px 0-15)
    v8f acc11 = acc00;                               // (co1 tile, px 16-31)

    // K ordered tap-major: k = tap*256 + ci  (blocks of 4 never straddle a tap)
    for (int tap = 0; tap < 9; ++tap) {
        const int ky = tap / 3, kx = tap % 3;
        const int xoff = ky * XCOL + kx + mn;               // Xs col index base
        const float* WA0 = Wc + ((size_t)(co0 + mn) * 256) * 9 + tap;
        const float* WA1 = Wc + ((size_t)(co1 + mn) * 256) * 9 + tap;
        for (int ci0 = 0; ci0 < 256; ci0 += 4) {
            const int cA = ci0 + half * 2;
            // A 16x4 layout (ISA 7.12.2): VGPR0 = K{0|2}, VGPR1 = K{1|3}
            v2f a0, a1, b0, b1;
            a0.x = WA0[(size_t)cA * 9];       a0.y = WA0[(size_t)(cA + 1) * 9];
            a1.x = WA1[(size_t)cA * 9];       a1.y = WA1[(size_t)(cA + 1) * 9];
            // B 4x16 mirrored layout: VGPR0 = K{0|2} rows, VGPR1 = K{1|3}
            const int bi = cA * XROW + xoff;
            b0.x = Xs[bi];            b0.y = Xs[bi + XROW];
            b1.x = Xs[bi + 16];       b1.y = Xs[bi + XROW + 16];
            acc00 = __builtin_amdgcn_wmma_f32_16x16x4_f32(false, a0, false, b0,
                                                          (short)0, acc00, false, false);
            acc01 = __builtin_amdgcn_wmma_f32_16x16x4_f32(false, a0, false, b1,
                                                          (short)0, acc01, false, false);
            acc10 = __builtin_amdgcn_wmma_f32_16x16x4_f32(false, a1, false, b0,
                                                          (short)0, acc10, false, false);
            acc11 = __builtin_amdgcn_wmma_f32_16x16x4_f32(false, a1, false, b1,
                                                          (short)0, acc11, false, false);
        }
    }
    __syncthreads();                         // done reading Xs -> reuse as h tile

    float* Hs = Xs;                          // 256 x 32, stride HSTR
    for (int r = 0; r < 8; ++r) {
        const int mm = r + half * 8;         // C/D layout: VGPR r -> M=r | M=r+8
        Hs[(co0 + mm) * HSTR + mn]      = fmaxf(acc00[r] + Bc[co0 + mm], 0.0f);
        Hs[(co0 + mm) * HSTR + mn + 16] = fmaxf(acc01[r] + Bc[co0 + mm], 0.0f);
        Hs[(co1 + mm) * HSTR + mn]      = fmaxf(acc10[r] + Bc[co1 + mm], 0.0f);
        Hs[(co1 + mm) * HSTR + mn + 16] = fmaxf(acc11[r] + Bc[co1 + mm], 0.0f);
    }
    __syncthreads();

    // 1x1 heads: 15 outputs x 32 pixels = 480 work items
    for (int t = tid; t < 15 * TPB; t += 256) {
        const int o = t / TPB, n2 = t % TPB;
        const float* wv; float bias;
        if (o < 12) { wv = Wreg + (size_t)o * 256;        bias = Breg[o];      }
        else        { wv = Wcls + (size_t)(o - 12) * 256; bias = Bcls[o - 12]; }
        float s = bias;
        for (int ci = 0; ci < 256; ++ci) s += wv[ci] * Hs[ci * HSTR + n2];
        const int gx = x0 + n2;
        if (gx < Wd) {
            const int pix = y * Wd + gx;
            if (o < 12)
                regOut[(size_t)b * N * 4 + ((size_t)pix * 3 + (o >> 2)) * 4 + (o & 3)] = s;
            else
                scrOut[(size_t)b * N + (size_t)pix * 3 + (o - 12)] = 1.0f / (1.0f + expf(-s));
        }
    }
}

// ------------------------- per-level exact top-k ----------------------------
__device__ __forceinline__ unsigned score_bin(unsigned key) {
    unsigned bv = key >> 19;                 // sigmoid >= 0 -> bits are monotone
    return bv > 2047u ? 2047u : bv;
}

__global__ void hist1K(const float* __restrict__ sc, int N, unsigned* __restrict__ hist) {
    for (int i = blockIdx.x * blockDim.x + threadIdx.x; i < N; i += gridDim.x * blockDim.x)
        atomicAdd(&hist[score_bin(__float_as_uint(sc[i]))], 1u);
}

__global__ void scan1K(const unsigned* __restrict__ hist, int N, int* __restrict__ res) {
    if (threadIdx.x) return;
    const int k = N < LVTOP ? N : LVTOP;
    unsigned cum = 0; int T1 = 0; unsigned m1 = 0; int found = 0;
    for (int bvi = 2047; bvi >= 0; --bvi) {
        unsigned c = hist[bvi];
        if (!found && cum + c >= (unsigned)k) { T1 = bvi; m1 = cum; found = 1; }
        if (!found) cum += c;
    }
    if (!found) { T1 = 0; m1 = cum; }        // N < k: degenerates to all-pass
    res[0] = T1; res[1] = (int)m1; res[2] = k;
}

__global__ void hist2K(const float* __restrict__ sc, int N,
                       const int* __restrict__ res, unsigned* __restrict__ hist2) {
    const int T1 = res[0];
    for (int i = blockIdx.x * blockDim.x + threadIdx.x; i < N; i += gridDim.x * blockDim.x) {
        unsigned key = __float_as_uint(sc[i]);
        if ((int)score_bin(key) == T1) atomicAdd(&hist2[(key >> 8) & 0x7FFu], 1u);
    }
}

__global__ void scan2K(const unsigned* __restrict__ hist2, int* __restrict__ res) {
    if (threadIdx.x) return;
    const int k = res[2];
    unsigned cum = (unsigned)res[1]; int T2 = 0; int found = 0;
    for (int s = 2047; s >= 0; --s) {
        unsigned c = hist2[s];
        if (!found && cum + c >= (unsigned)k) { T2 = s; found = 1; }
        if (!found) cum += c;
    }
    res[3] = found ? T2 : 0;
}

__global__ void compactK(const float* __restrict__ sc, int N, const int* __restrict__ res,
                         unsigned* __restrict__ cnt, float* __restrict__ candS,
                         int* __restrict__ candI) {
    const int T1 = res[0], T2 = res[3];
    for (int i = blockIdx.x * blockDim.x + threadIdx.x; i < N; i += gridDim.x * blockDim.x) {
        unsigned key = __float_as_uint(sc[i]);
        int bvi = (int)score_bin(key);
        int sub = (int)((key >> 8) & 0x7FFu);
        if (bvi > T1 || (bvi == T1 && sub >= T2)) {
            unsigned p = atomicAdd(cnt, 1u);
            if (p < CAND_CAP) { candS[p] = sc[i]; candI[p] = i; }
        }
    }
}

// exact rank among candidates: (score desc, index asc) == jax.lax.top_k order
__global__ __launch_bounds__(1024)
void selectK(const unsigned* __restrict__ cnt, const float* __restrict__ candS,
             const int* __restrict__ candI, const int* __restrict__ res,
             float* __restrict__ topv, int* __restrict__ topi) {
    __shared__ float ss[CAND_CAP];
    __shared__ int   si[CAND_CAP];
    int C = (int)*cnt; if (C > CAND_CAP) C = CAND_CAP;
    const int k = res[2];
    for (int t = threadIdx.x; t < C; t += 1024) { ss[t] = candS[t]; si[t] = candI[t]; }
    __syncthreads();
    for (int i = threadIdx.x; i < C; i += 1024) {
        const float s = ss[i]; const int id = si[i];
        int rank = 0;
        for (int j = 0; j < C; ++j) {
            float t = ss[j];
            rank += (t > s) || (t == s && si[j] < id);
        }
        if (rank < k) { topv[rank] = s; topi[rank] = id; }
    }
}

// ------------------------- decode + clip + minsize --------------------------
__global__ void decodeK(const float* __restrict__ topv, const int* __restrict__ topi,
                        const float* __restrict__ reg, float* __restrict__ boxes,
                        float* __restrict__ sc, int Wd, float stride, float asize) {
    const int r = blockIdx.x * blockDim.x + threadIdx.x;
    if (r >= LVTOP) return;
    const float v = topv[r];
    if (v <= -0.5f) return;                  // padded slot: boxes stay 0, sc -1
    const int i = topi[r];
    const int a = i % 3, pix = i / 3;
    const int yy = pix / Wd, xx = pix % Wd;
    const float ratio = (a == 0) ? 2.0f : (a == 1 ? 1.0f : 0.5f);
    const float hr  = sqrtf(ratio);
    const float wsz = asize / hr, hsz = asize * hr;
    const float bx0 = rintf(-0.5f * wsz), by0 = rintf(-0.5f * hsz);
    const float bx1 = rintf( 0.5f * wsz), by1 = rintf( 0.5f * hsz);
    const float sx = (float)xx * stride, sy = (float)yy * stride;
    const float px0 = sx + bx0, py0 = sy + by0, px1 = sx + bx1, py1 = sy + by1;
    const float pw = px1 - px0, ph = py1 - py0;
    const float pcx = px0 + 0.5f * pw, pcy = py0 + 0.5f * ph;
    const float* rg = reg + (size_t)i * 4;
    const float dx = rg[0], dy = rg[1];
    const float dw = fminf(rg[2], BBOX_CLIP), dh = fminf(rg[3], BBOX_CLIP);
    const float cx = dx * pw + pcx, cy = dy * ph + pcy;
    const float w = expf(dw) * pw, h = expf(dh) * ph;
    const float X0 = fminf(fmaxf(cx - 0.5f * w, 0.f), IMG_W_F);
    const float Y0 = fminf(fmaxf(cy - 0.5f * h, 0.f), IMG_H_F);
    const float X1 = fminf(fmaxf(cx + 0.5f * w, 0.f), IMG_W_F);
    const float Y1 = fminf(fmaxf(cy + 0.5f * h, 0.f), IMG_H_F);
    boxes[(size_t)r * 4 + 0] = X0; boxes[(size_t)r * 4 + 1] = Y0;
    boxes[(size_t)r * 4 + 2] = X1; boxes[(size_t)r * 4 + 3] = Y1;
    sc[r] = ((X1 - X0) > 1e-3f && (Y1 - Y0) > 1e-3f) ? v : -1.0f;
}

// --------------------------- greedy per-level NMS ---------------------------
__global__ __launch_bounds__(1024)
void nmsK(const float* __restrict__ boxes, float* __restrict__ sc) {
    __shared__ float bx[LVTOP * 4];
    __shared__ float ar[LVTOP];
    __shared__ int   kp[LVTOP];
    const int tid = threadIdx.x;
    for (int t = tid; t < LVTOP * 4; t += 1024) bx[t] = boxes[t];
    __syncthreads();
    if (tid < LVTOP) {
        ar[tid] = (bx[tid*4+2] - bx[tid*4+0]) * (bx[tid*4+3] - bx[tid*4+1]);
        kp[tid] = sc[tid] > -0.5f ? 1 : 0;
    }
    __syncthreads();
    for (int i = 0; i < LVTOP; ++i) {
        if (kp[i]) {                         // uniform LDS read
            const int j = tid;
            if (j > i && j < LVTOP && kp[j]) {
                const float lx = fmaxf(bx[i*4+0], bx[j*4+0]);
                const float ly = fmaxf(bx[i*4+1], bx[j*4+1]);
                const float rx = fminf(bx[i*4+2], bx[j*4+2]);
                const float ry = fminf(bx[i*4+3], bx[j*4+3]);
                const float iw = fmaxf(rx - lx, 0.f), ih = fmaxf(ry - ly, 0.f);
                const float inter = iw * ih;
                const float iou = inter / fmaxf(ar[i] + ar[j] - inter, 1e-9f);
                if (iou > 0.7f) kp[j] = 0;
            }
        }
        __syncthreads();
    }
    if (tid < LVTOP) sc[tid] = kp[tid] ? sc[tid] : -1.0f;
}

// --------------------------- image-level top-1000 ---------------------------
__global__ __launch_bounds__(1024)
void finalK(const float* __restrict__ scAll, const float* __restrict__ boxAll,
            int b, float* __restrict__ out) {
    __shared__ float s[5 * LVTOP];
    for (int t = threadIdx.x; t < 5 * LVTOP; t += 1024) {
        const int l = t / LVTOP, r = t % LVTOP;
        s[t] = scAll[(size_t)(l * 2 + b) * LVTOP + r];
    }
    __syncthreads();
    for (int g = threadIdx.x; g < 5 * LVTOP; g += 1024) {
        const float v = s[g];
        int rank = 0;
        for (int j = 0; j < 5 * LVTOP; ++j) {
            float t = s[j];
            rank += (t > v) || (t == v && j < g);
        }
        if (rank < 1000) {
            const int l = g / LVTOP, r = g % LVTOP;
            const float* bs = boxAll + ((size_t)(l * 2 + b) * LVTOP + r) * 4;
            float* dst = out + ((size_t)b * 1000 + rank) * 5;
            dst[0] = bs[0]; dst[1] = bs[1]; dst[2] = bs[2]; dst[3] = bs[3]; dst[4] = v;
        }
    }
}

// ------------------------------- launcher -----------------------------------
extern "C" void kernel_launch(void* const* d_in, const int* in_sizes, int n_in,
                              void* d_out, int out_size, void* d_ws, size_t ws_size,
                              hipStream_t stream)
{
    (void)in_sizes; (void)n_in; (void)out_size; (void)ws_size;
    const float* P[5] = {(const float*)d_in[0], (const float*)d_in[1],
                         (const float*)d_in[2], (const float*)d_in[3],
                         (const float*)d_in[4]};
    const float* Wc   = (const float*)d_in[5];
    const float* Bc   = (const float*)d_in[6];
    const float* Wcls = (const float*)d_in[7];
    const float* Bcls = (const float*)d_in[8];
    const float* Wreg = (const float*)d_in[9];
    const float* Breg = (const float*)d_in[10];
    float* out = (float*)d_out;

    static const int LH[5] = {200, 100, 50, 25, 13};
    static const int LW[5] = {304, 152, 76, 38, 19};
    static const int LS[5] = {4, 8, 16, 32, 64};
    static const int LZ[5] = {32, 64, 128, 256, 512};
    int LN[5]; size_t scrOff[5], regOff[5];
    size_t so = 0, ro = 0;
    for (int l = 0; l < 5; ++l) {
        LN[l] = LH[l] * LW[l] * 3;
        scrOff[l] = so; regOff[l] = ro;
        so += (size_t)2 * LN[l];
        ro += (size_t)8 * LN[l];
    }

    char* wp = (char*)d_ws;
    size_t off = 0;
    auto take = [&](size_t bytes) -> char* {
        char* p = wp + off;
        off = (off + bytes + 255) & ~(size_t)255;
        return p;
    };
    float*    scrAll = (float*)take(so * 4);
    float*    regAll = (float*)take(ro * 4);
    unsigned* hist1  = (unsigned*)take((size_t)10 * 2048 * 4);
    unsigned* hist2  = (unsigned*)take((size_t)10 * 2048 * 4);
    int*      scanR  = (int*)take((size_t)10 * 8 * 4);
    unsigned* cCnt   = (unsigned*)take((size_t)10 * 4);
    float*    candS  = (float*)take((size_t)10 * CAND_CAP * 4);
    int*      candI  = (int*)take((size_t)10 * CAND_CAP * 4);
    float*    topv   = (float*)take((size_t)10 * LVTOP * 4);
    int*      topi   = (int*)take((size_t)10 * LVTOP * 4);
    float*    boxes  = (float*)take((size_t)10 * LVTOP * 4 * 4);
    float*    scBuf  = (float*)take((size_t)10 * LVTOP * 4);

    // re-init control state every call (ws is not re-poisoned between replays)
    fill_u32<<<(10*2048 + 255)/256, 256, 0, stream>>>(hist1, 10*2048);
    fill_u32<<<(10*2048 + 255)/256, 256, 0, stream>>>(hist2, 10*2048);
    fill_u32<<<1, 128, 0, stream>>>((unsigned*)scanR, 80);
    fill_u32<<<1, 32, 0, stream>>>(cCnt, 10);
    fill_f32<<<(10*LVTOP + 255)/256, 256, 0, stream>>>(topv, 10*LVTOP, -1.0f);
    fill_u32<<<(10*LVTOP + 255)/256, 256, 0, stream>>>((unsigned*)topi, 10*LVTOP);
    fill_f32<<<(10*LVTOP*4 + 255)/256, 256, 0, stream>>>(boxes, 10*LVTOP*4, 0.0f);
    fill_f32<<<(10*LVTOP + 255)/256, 256, 0, stream>>>(scBuf, 10*LVTOP, -1.0f);

    // fused conv3x3+relu+heads per level (WMMA fp32)
    for (int l = 0; l < 5; ++l) {
        dim3 grid((LW[l] + TPB - 1) / TPB, LH[l], 2);
        conv_head_fused<<<grid, 256, 0, stream>>>(P[l], Wc, Bc, Wcls, Bcls, Wreg, Breg,
                                                  regAll + regOff[l], scrAll + scrOff[l],
                                                  LH[l], LW[l], LN[l]);
    }

    // per-(level,batch) exact top-1000 via radix threshold + rank selection
    for (int l = 0; l < 5; ++l)
        for (int b = 0; b < 2; ++b) {
            const int inst = l * 2 + b;
            const float* scp = scrAll + scrOff[l] + (size_t)b * LN[l];
            hist1K<<<64, 256, 0, stream>>>(scp, LN[l], hist1 + inst * 2048);
        }
    for (int l = 0; l < 5; ++l)
        for (int b = 0; b < 2; ++b) {
            const int inst = l * 2 + b;
            scan1K<<<1, 32, 0, stream>>>(hist1 + inst * 2048, LN[l], scanR + inst * 8);
        }
    for (int l = 0; l < 5; ++l)
        for (int b = 0; b < 2; ++b) {
            const int inst = l * 2 + b;
            const float* scp = scrAll + scrOff[l] + (size_t)b * LN[l];
            hist2K<<<64, 256, 0, stream>>>(scp, LN[l], scanR + inst * 8, hist2 + inst * 2048);
        }
    for (int l = 0; l < 5; ++l)
        for (int b = 0; b < 2; ++b) {
            const int inst = l * 2 + b;
            scan2K<<<1, 32, 0, stream>>>(hist2 + inst * 2048, scanR + inst * 8);
        }
    for (int l = 0; l < 5; ++l)
        for (int b = 0; b < 2; ++b) {
            const int inst = l * 2 + b;
            const float* scp = scrAll + scrOff[l] + (size_t)b * LN[l];
            compactK<<<64, 256, 0, stream>>>(scp, LN[l], scanR + inst * 8, cCnt + inst,
                                             candS + (size_t)inst * CAND_CAP,
                                             candI + (size_t)inst * CAND_CAP);
        }
    for (int l = 0; l < 5; ++l)
        for (int b = 0; b < 2; ++b) {
            const int inst = l * 2 + b;
            selectK<<<1, 1024, 0, stream>>>(cCnt + inst,
                                            candS + (size_t)inst * CAND_CAP,
                                            candI + (size_t)inst * CAND_CAP,
                                            scanR + inst * 8,
                                            topv + (size_t)inst * LVTOP,
                                            topi + (size_t)inst * LVTOP);
        }
    for (int l = 0; l < 5; ++l)
        for (int b = 0; b < 2; ++b) {
            const int inst = l * 2 + b;
            decodeK<<<(LVTOP + 255)/256, 256, 0, stream>>>(
                topv + (size_t)inst * LVTOP, topi + (size_t)inst * LVTOP,
                regAll + regOff[l] + (size_t)b * LN[l] * 4,
                boxes + (size_t)inst * LVTOP * 4, scBuf + (size_t)inst * LVTOP,
                LW[l], (float)LS[l], (float)LZ[l]);
        }
    for (int l = 0; l < 5; ++l)
        for (int b = 0; b < 2; ++b) {
            const int inst = l * 2 + b;
            nmsK<<<1, 1024, 0, stream>>>(boxes + (size_t)inst * LVTOP * 4,
                                         scBuf + (size_t)inst * LVTOP);
        }
    for (int b = 0; b < 2; ++b)
        finalK<<<1, 1024, 0, stream>>>(scBuf, boxes, b, out);
}